// LaplacianNetwork_49761491091692
// MI455X (gfx1250) — compile-verified
//
#include <hip/hip_runtime.h>
#include <hip/hip_bf16.h>

#define JNT 16
#define DLAT 10
#define HDIM 64
#define PVERT 49152

typedef float v2f __attribute__((ext_vector_type(2)));
typedef float v8f __attribute__((ext_vector_type(8)));

// ---------------------------------------------------------------------------
// Kernel 1: compute dh[j][h] = (relu(feat.w1+b1) - relu(zfeat.w1+b1)) / 1000
// Tiny (16x64 outputs); one block.
// ---------------------------------------------------------------------------
__global__ void lapnet_compute_dh(const float* __restrict__ x,
                                  const float* __restrict__ lat,
                                  const float* __restrict__ w1,
                                  const float* __restrict__ b1,
                                  const int*   __restrict__ neigh,
                                  const float* __restrict__ nmask,
                                  float* __restrict__ dh) {
  int t = threadIdx.x;
  for (int idx = t; idx < JNT * HDIM; idx += 256) {
    int j = idx >> 6;
    const float* w = w1 + (size_t)idx * 37;
    // latent-only part (shared between feat and zfeat)
    float a_lat = 0.f;
    #pragma unroll
    for (int d = 0; d < DLAT; ++d) a_lat += w[27 + d] * lat[j * DLAT + d];
    // neighbor-gathered pose part
    float a_full = a_lat;
    #pragma unroll
    for (int k = 0; k < 3; ++k) {
      int   src = neigh[j * 3 + k];
      float m   = nmask[j * 3 + k];
      const float* pose = x + src * 9;
      #pragma unroll
      for (int i = 0; i < 9; ++i) a_full += w[k * 9 + i] * (pose[i] * m);
    }
    float b  = b1[idx];
    float hf = fmaxf(a_full + b, 0.f);
    float h0 = fmaxf(a_lat  + b, 0.f);
    dh[idx] = (hf - h0) * 1e-3f;   // fold the /1000 here (everything after is linear)
  }
}

// ---------------------------------------------------------------------------
// Kernel 2: out[o] = iv[o] + sum_j relu(mask[j,p]) * (dh[j,:] . w2[j,o,:])
// One wave per 16-output tile. WMMA f32 16x16x4 with broadcast-A (dh) and
// B = 16 rows of w2. Lane n ends with the tile's output n in c[0].
// ---------------------------------------------------------------------------
__global__ void lapnet_main(const float* __restrict__ w2,
                            const float* __restrict__ maskp,
                            const float* __restrict__ iv,
                            const float* __restrict__ dh_g,
                            float* __restrict__ out) {
  __shared__ float dh_s[JNT * HDIM];
  const int tid = threadIdx.x;
  for (int i = tid; i < JNT * HDIM; i += 256) dh_s[i] = dh_g[i];
  __syncthreads();

  const int lane  = tid & 31;
  const int n     = lane & 15;          // column / output-in-tile
  const int half2 = (lane >> 4) << 1;   // lanes 0-15 -> K{0,1}; lanes 16-31 -> K{2,3}
  const int wave      = blockIdx.x * 8 + (tid >> 5);
  const int waveCount = gridDim.x * 8;
  const int nTiles    = (3 * PVERT) / 16;   // 9216

  for (int tile = wave; tile < nTiles; tile += waveCount) {
    const int o = tile * 16 + n;
    const int p = o / 3;
    float acc = 0.f;
    #pragma unroll 1
    for (int j = 0; j < JNT; ++j) {
      const float* w2t = w2 + ((size_t)j * (3 * PVERT) + (size_t)tile * 16) * HDIM;
      // warm L2 for next joint's 4KB tile region (global_prefetch_b8, 1 line/lane)
      if (j + 1 < JNT)
        __builtin_prefetch(w2t + (size_t)(3 * PVERT) * HDIM + lane * 32, 0, 1);
      const float* bp = w2t + n * HDIM + half2;   // B operand: row o_n, K-halves split
      const float* ap = dh_s + j * HDIM + half2;  // A operand: dh broadcast across M
      v8f c = {};
      #pragma unroll
      for (int hb = 0; hb < HDIM; hb += 4) {
        v2f b = *(const v2f*)(bp + hb);   // global_load_b64, each w2 byte read once
        v2f a = *(const v2f*)(ap + hb);   // ds_load_b64 from staged dh
        c = __builtin_amdgcn_wmma_f32_16x16x4_f32(
            /*neg_a=*/false, a, /*neg_b=*/false, b,
            /*c_mod=*/(short)0, c, /*reuse_a=*/false, /*reuse_b=*/false);
      }
      float m = maskp[(size_t)j * PVERT + p];
      m = m > 0.f ? m : 0.f;              // relu(mask)
      acc = fmaf(m, c[0], acc);           // c[0] == dot(dh_j, w2_j[o_n,:])
    }
    if (lane < 16) out[o] = iv[o] + acc;  // coalesced: lane n <-> output o_base+n
  }
}

// ---------------------------------------------------------------------------
// Launch
// ---------------------------------------------------------------------------
extern "C" void kernel_launch(void* const* d_in, const int* in_sizes, int n_in,
                              void* d_out, int out_size, void* d_ws, size_t ws_size,
                              hipStream_t stream) {
  const float* x     = (const float*)d_in[0];
  const float* iv    = (const float*)d_in[1];
  const float* lat   = (const float*)d_in[2];
  const float* maskp = (const float*)d_in[3];
  const float* w1    = (const float*)d_in[4];
  const float* b1    = (const float*)d_in[5];
  const float* w2    = (const float*)d_in[6];
  // d_in[7] = b2: cancels in corr - corr0
  const int*   neigh = (const int*)d_in[8];
  const float* nmask = (const float*)d_in[9];
  float* out = (float*)d_out;
  float* dh  = (float*)d_ws;   // 16*64 floats

  lapnet_compute_dh<<<1, 256, 0, stream>>>(x, lat, w1, b1, neigh, nmask, dh);
  // 9216 tiles, 8 waves/block -> 1152 blocks = one tile per wave
  lapnet_main<<<1152, 256, 0, stream>>>(w2, maskp, iv, dh, out);
}